// CoupleLoss_88656714924199
// MI455X (gfx1250) — compile-verified
//
#include <hip/hip_runtime.h>

typedef __attribute__((ext_vector_type(2))) float v2f;
typedef __attribute__((ext_vector_type(8))) float v8f;

#define NUM_IDS 100000
#define FEAT    512
#define BATCH   512
#define KNEG    100
#define MARGIN  0.03f
#define NTHREADS 224   // 7 waves of 32: one wave per 16-wide k-tile (7*16 = 112 >= 100)

// ---------------------------------------------------------------------------
// Kernel 1: init inverse scatter map to -1
// ---------------------------------------------------------------------------
__global__ void cl_init_map(int* __restrict__ map) {
    int i = blockIdx.x * blockDim.x + threadIdx.x;
    if (i < NUM_IDS) map[i] = -1;
}

// ---------------------------------------------------------------------------
// Kernel 2: map[label[b]] = max b  (deterministic "last write wins" stand-in
// for protos.at[label].set(teachor_ftr) with duplicate labels)
// ---------------------------------------------------------------------------
__global__ void cl_scatter_map(const int* __restrict__ label, int* __restrict__ map) {
    int b = blockIdx.x * blockDim.x + threadIdx.x;
    if (b < BATCH) atomicMax(&map[label[b]], b);
}

// ---------------------------------------------------------------------------
// Kernel 3: per-b loss partials via V_WMMA_F32_16X16X4_F32.
// One block per batch element b. 7 waves; wave w owns k-tile [16w, 16w+16).
// A (16x4 f32): 16 gathered rows, 4 consecutive features.
//   lanes 0-15: row m=lane, K=0,1 ; lanes 16-31: row m=lane-16, K=2,3.
// B (4x16 f32): diff[b] in column N=0 only. Column masking is done ONCE by
//   pointer selection: lanes 0/16 read from sdiff, all other lanes read from
//   an identically-sized zeroed LDS region -> inner loop has no cndmasks.
// D column 0 = 16 dot products: lane 0 holds M=0..7 (vgpr r), lane 16 M=8..15.
// ---------------------------------------------------------------------------
__global__ __launch_bounds__(NTHREADS)
void cl_compute(const float* __restrict__ ftr,
                const float* __restrict__ tea,
                const int*   __restrict__ label,
                const float* __restrict__ protos,
                const int*   __restrict__ idH,
                const int*   __restrict__ map,
                float*       __restrict__ partial) {
    __shared__ float sdiff[FEAT];
    __shared__ float szero[FEAT];
    __shared__ float wsum[14];

    const int b   = blockIdx.x;
    const int tid = threadIdx.x;

    // diff[b] = ftr[b] - teachor[b] staged in LDS; zero page for masked lanes
    for (int i = tid; i < FEAT; i += NTHREADS) {
        sdiff[i] = ftr[(size_t)b * FEAT + i] - tea[(size_t)b * FEAT + i];
        szero[i] = 0.0f;
    }
    __syncthreads();

    const int wave = tid >> 5;        // 0..6  -> k-tile index
    const int lane = tid & 31;
    const int m    = lane & 15;       // row within tile
    const int half = lane >> 4;       // 0: K=0,1  1: K=2,3
    const int k    = wave * 16 + m;

    // gather row base, substituting teachor rows for scattered ids
    const int lb  = label[b];
    int gid = 0;
    if (k < KNEG) gid = idH[(size_t)lb * KNEG + k];
    const int owner = map[gid];
    const float* rowbase = (k < KNEG && owner >= 0)
                               ? (tea    + (size_t)owner * FEAT)
                               : (protos + (size_t)gid   * FEAT);
    __builtin_prefetch(rowbase, 0, 0);

    // A: float2 view; lane reads rowbase[d + 2*half .. +1]
    const v2f* rp2 = (const v2f*)rowbase;
    // B: per-lane base pointer selected once (no per-iteration masking)
    const float* bbase = ((m == 0) ? sdiff : szero) + half * 2;

    v8f acc0 = {}, acc1 = {}, acc2 = {}, acc3 = {};
    for (int d = 0; d < FEAT; d += 16) {
        v2f a0 = rp2[((d     ) >> 1) + half];
        v2f a1 = rp2[((d +  4) >> 1) + half];
        v2f a2 = rp2[((d +  8) >> 1) + half];
        v2f a3 = rp2[((d + 12) >> 1) + half];

        v2f b0 = { bbase[d     ], bbase[d +  1] };
        v2f b1 = { bbase[d +  4], bbase[d +  5] };
        v2f b2 = { bbase[d +  8], bbase[d +  9] };
        v2f b3 = { bbase[d + 12], bbase[d + 13] };

        acc0 = __builtin_amdgcn_wmma_f32_16x16x4_f32(false, a0, false, b0,
                                                     (short)0, acc0, false, false);
        acc1 = __builtin_amdgcn_wmma_f32_16x16x4_f32(false, a1, false, b1,
                                                     (short)0, acc1, false, false);
        acc2 = __builtin_amdgcn_wmma_f32_16x16x4_f32(false, a2, false, b2,
                                                     (short)0, acc2, false, false);
        acc3 = __builtin_amdgcn_wmma_f32_16x16x4_f32(false, a3, false, b3,
                                                     (short)0, acc3, false, false);
    }

    // Extract column 0: lane 0 -> M=0..7, lane 16 -> M=8..15
    if (m == 0) {
        float local = 0.0f;
        for (int r = 0; r < 8; ++r) {
            const int kk = wave * 16 + half * 8 + r;
            if (kk < KNEG) {
                float v = (acc0[r] + acc1[r]) + (acc2[r] + acc3[r]) - MARGIN;
                local += (v > 0.0f) ? v : 0.0f;
            }
        }
        wsum[wave * 2 + half] = local;
    }
    __syncthreads();

    if (tid == 0) {
        float s = 0.0f;
        for (int i = 0; i < 14; ++i) s += wsum[i];
        partial[b] = s;
    }
}

// ---------------------------------------------------------------------------
// Kernel 4: deterministic tree reduction of 512 partials -> mean
// ---------------------------------------------------------------------------
__global__ void cl_reduce(const float* __restrict__ partial, float* __restrict__ out) {
    __shared__ float s[BATCH];
    const int t = threadIdx.x;
    s[t] = partial[t];
    __syncthreads();
    for (int o = BATCH / 2; o > 0; o >>= 1) {
        if (t < o) s[t] += s[t + o];
        __syncthreads();
    }
    if (t == 0) out[0] = s[0] * (1.0f / (float)(BATCH * KNEG));
}

// ---------------------------------------------------------------------------
extern "C" void kernel_launch(void* const* d_in, const int* in_sizes, int n_in,
                              void* d_out, int out_size, void* d_ws, size_t ws_size,
                              hipStream_t stream) {
    const float* ftr    = (const float*)d_in[0];
    const float* tea    = (const float*)d_in[1];
    const int*   label  = (const int*)d_in[2];
    const float* protos = (const float*)d_in[3];
    const int*   idH    = (const int*)d_in[4];
    float*       out    = (float*)d_out;

    float* partial = (float*)d_ws;                       // 512 floats
    int*   map     = (int*)((char*)d_ws + 4096);         // NUM_IDS ints (~400 KB)

    cl_init_map   <<<(NUM_IDS + 255) / 256, 256, 0, stream>>>(map);
    cl_scatter_map<<<(BATCH + 255) / 256, 256, 0, stream>>>(label, map);
    cl_compute    <<<BATCH, NTHREADS, 0, stream>>>(ftr, tea, label, protos, idH, map, partial);
    cl_reduce     <<<1, BATCH, 0, stream>>>(partial, out);
}